// C3S_RegularLoss_86526411145660
// MI455X (gfx1250) — compile-verified
//
#include <hip/hip_runtime.h>

// ---------------------------------------------------------------------------
// C3S regularizer loss on MI455X (gfx1250).
//
// x: (8192, 8192) f32, viewed as (B=8192, P=4, D=2048).
//  Pass 1 (bandwidth bound, 256 MB @ 23.3 TB/s ~ 11us):
//    per (row, part): ssq reduce -> rsqrt -> scale -> accumulate column sums
//    into sacc[4][2048] (f32 global atomics, ~1M total).
//    Register double-buffered: row r+1's global_load_b128 pair is issued
//    before row r's reduction/barrier, so HBM latency overlaps the reduce.
//    One barrier per row: every thread sums the 8 per-wave LDS partials
//    itself (parity double-buffered), no thread-0 broadcast round trip.
//  Pass 2 (tiny): s = sacc / B; corr = s s^T via V_WMMA_F32_16X16X4_F32
//    (f32 WMMA required: s entries ~2e-6, corr ~1e-8 -> f16 would underflow),
//    K=2048 split over 8 waves, partial 4x4s reduced in LDS, scalar loss out.
// ---------------------------------------------------------------------------

#define NPARTS 4
#define PART_D 2048
#define ROWLEN (NPARTS * PART_D)   // 8192 floats per row
#define TPB 256
#define NWAVES (TPB / 32)
#define ROWS_PER_BLOCK 64

typedef __attribute__((ext_vector_type(2))) float v2f;
typedef __attribute__((ext_vector_type(8))) float v8f;

__global__ void zero_ws_kernel(float* __restrict__ p, int n) {
  int i = blockIdx.x * blockDim.x + threadIdx.x;
  if (i < n) p[i] = 0.0f;
}

// Grid: (rows / ROWS_PER_BLOCK, NPARTS). Block: 256 threads (8 waves).
// Thread t owns columns [4t..4t+3] and [1024+4t..1024+4t+3] of its part.
__global__ __launch_bounds__(TPB)
void norm_accum_kernel(const float* __restrict__ x, float* __restrict__ sacc) {
  const int part = blockIdx.y;
  const int row0 = blockIdx.x * ROWS_PER_BLOCK;
  const int t    = threadIdx.x;
  const int wave = t >> 5;
  const int lane = t & 31;

  __shared__ float red[2][NWAVES];   // parity double buffer -> 1 barrier/row

  float acc[8];
#pragma unroll
  for (int i = 0; i < 8; ++i) acc[i] = 0.0f;

  const float* base = x + (size_t)row0 * ROWLEN + (size_t)part * PART_D
                        + (size_t)(4 * t);

  // Process one row held in (v0, v1); single __syncthreads per row.
  auto process = [&](const float4& v0, const float4& v1, int parity) {
    float ssq = v0.x*v0.x + v0.y*v0.y + v0.z*v0.z + v0.w*v0.w
              + v1.x*v1.x + v1.y*v1.y + v1.z*v1.z + v1.w*v1.w;
#pragma unroll
    for (int off = 16; off >= 1; off >>= 1) ssq += __shfl_xor(ssq, off, 32);
    if (lane == 0) red[parity][wave] = ssq;
    __syncthreads();
    float tot = 0.0f;
#pragma unroll
    for (int w = 0; w < NWAVES; ++w) tot += red[parity][w];  // LDS broadcast
    const float rn = rsqrtf(tot);
    acc[0] += v0.x * rn; acc[1] += v0.y * rn;
    acc[2] += v0.z * rn; acc[3] += v0.w * rn;
    acc[4] += v1.x * rn; acc[5] += v1.y * rn;
    acc[6] += v1.z * rn; acc[7] += v1.w * rn;
  };

  // Preload row 0.
  float4 v0 = *(const float4*)(base);            // global_load_b128
  float4 v1 = *(const float4*)(base + 1024);

  for (int r = 0; r < ROWS_PER_BLOCK - 1; ++r) {
    const float* pn = base + (size_t)(r + 1) * ROWLEN;
    // Issue next row's loads BEFORE this row's reduction/barrier.
    const float4 n0 = *(const float4*)(pn);
    const float4 n1 = *(const float4*)(pn + 1024);
    __builtin_prefetch(pn + ROWLEN, 0, 0);       // global_prefetch_b8 (r+2)
    __builtin_prefetch(pn + ROWLEN + 1024, 0, 0);

    process(v0, v1, r & 1);
    v0 = n0; v1 = n1;
  }
  process(v0, v1, (ROWS_PER_BLOCK - 1) & 1);

  float* sp = sacc + part * PART_D + 4 * t;
#pragma unroll
  for (int i = 0; i < 4; ++i) atomicAdd(&sp[i], acc[i]);         // global_atomic_add_f32
#pragma unroll
  for (int i = 0; i < 4; ++i) atomicAdd(&sp[1024 + i], acc[4 + i]);
}

// One block of 256 threads = 8 waves; wave w covers K in [w*256, w*256+256).
// A (16x4 f32) layout: lane -> M (lanes 0-15 / 16-31 mirror), VGPR0/1 -> K
// {2h, 2h+1} with h = lane>>4.  B (4x16) layout is the lane->N mirror with the
// same K striping, so for a Gram matrix the SAME v2f serves as A and B.
#define FIN_TPB 256
#define KSLICE  (PART_D / (FIN_TPB / 32))   // 256 K-values per wave

__global__ __launch_bounds__(FIN_TPB)
void corr_loss_kernel(const float* __restrict__ sacc,
                      const float* __restrict__ gamma,
                      float* __restrict__ out, float inv_rows) {
  const int t    = threadIdx.x;
  const int wave = t >> 5;
  const int lane = t & 31;
  const int half = lane >> 4;
  const int m    = lane & 15;
  // Rows >= NPARTS contribute zeros (select via cndmask, no EXEC divergence).
  const float scale = (m < NPARTS) ? inv_rows : 0.0f;
  const int   mrow  = (m < NPARTS) ? m : 0;
  const float* rowp = sacc + mrow * PART_D + 2 * half;
  const int k_begin = wave * KSLICE;

  v8f c = {};
  for (int k0 = k_begin; k0 < k_begin + KSLICE; k0 += 4) {
    v2f a;
    a.x = rowp[k0]     * scale;   // s[m][k0 + 2h]
    a.y = rowp[k0 + 1] * scale;   // s[m][k0 + 2h + 1]
    // D = A x B + C, A == B  ->  partial Gram matrix of s rows.
    c = __builtin_amdgcn_wmma_f32_16x16x4_f32(
        /*neg_a=*/false, a, /*neg_b=*/false, a,
        /*c_mod=*/(short)0, c, /*reuse_a=*/false, /*reuse_b=*/false);
  }

  // C/D layout: VGPR i holds row M=i for lanes 0-15 (N = lane).
  __shared__ float pc[FIN_TPB / 32][NPARTS][NPARTS];
  if (lane < NPARTS) {
#pragma unroll
    for (int i = 0; i < NPARTS; ++i) pc[wave][i][lane] = c[i];
  }
  __syncthreads();
  if (t == 0) {
    float total = 0.0f, trace = 0.0f;
    for (int w = 0; w < FIN_TPB / 32; ++w)
      for (int i = 0; i < NPARTS; ++i) {
        trace += pc[w][i][i];
        for (int j = 0; j < NPARTS; ++j) total += pc[w][i][j];
      }
    out[0] = (total - 3.0f * trace + 2.0f * (float)NPARTS) * 0.5f * gamma[0];
  }
}

extern "C" void kernel_launch(void* const* d_in, const int* in_sizes, int n_in,
                              void* d_out, int out_size, void* d_ws, size_t ws_size,
                              hipStream_t stream) {
  const float* x     = (const float*)d_in[0];
  const float* gamma = (const float*)d_in[1];
  float* out  = (float*)d_out;
  float* sacc = (float*)d_ws;          // 4*2048 f32 = 32 KB scratch

  const int rows = in_sizes[0] / ROWLEN;   // 8192
  const int nacc = NPARTS * PART_D;

  zero_ws_kernel<<<(nacc + 255) / 256, 256, 0, stream>>>(sacc, nacc);

  dim3 grid(rows / ROWS_PER_BLOCK, NPARTS);
  norm_accum_kernel<<<grid, TPB, 0, stream>>>(x, sacc);

  corr_loss_kernel<<<1, FIN_TPB, 0, stream>>>(sacc, gamma, out,
                                              1.0f / (float)rows);
}